// rtn_gen_2138893713926
// MI455X (gfx1250) — compile-verified
//
#include <hip/hip_runtime.h>
#include <cstddef>

// ---------------------------------------------------------------------------
// 3-layer fused LSTM (B=4096, T=512, D=H=16) + residual, f32, for gfx1250.
// One wave32 owns a 16-row batch tile and runs all 3 layers over all T.
// Recurrent + input projections done with V_WMMA_F32_16X16X4_F32 (exact f32):
//   gates[16b x 64g] = bias + x_t[16x16] @ Wih^T + h[16x16] @ Whh^T
//   -> per timestep: 4 gate N-tiles x (4 x-WMMAs + 4 h-WMMAs) = 32 WMMAs.
// Gate nonlinearities use hardware v_tanh_f32 when available.
// Intermediate layer outputs ping through d_out in place (per-tile private).
// ---------------------------------------------------------------------------

typedef __attribute__((ext_vector_type(2))) float v2f;
typedef __attribute__((ext_vector_type(8))) float v8f;

#define B_TOT 4096
#define T_TOT 512
#define HD    16            // input dim == hidden dim
#define TC    16            // timesteps staged per LDS chunk
#define NCH   (T_TOT / TC)  // 32 chunks
#define RP    18            // padded LDS row pitch in floats (conflict-free, 8B aligned pairs)
#define WAVES 2             // waves per workgroup (independent batch tiles)

#define CHUNK_F (TC * 16 * RP)   // 4608 floats: [t][row][d(+pad)]
#define HBUF_F  (16 * RP)        // 288 floats:  chunk-boundary h tile
#define WAVE_F  (CHUNK_F + HBUF_F)

#if __has_builtin(__builtin_amdgcn_tanhf)
// gfx1250 hardware transcendental: v_tanh_f32
__device__ __forceinline__ float tanh_(float x) {
    return __builtin_amdgcn_tanhf(x);
}
__device__ __forceinline__ float sigm_(float x) {
    return __builtin_amdgcn_tanhf(x * 0.5f) * 0.5f + 0.5f;
}
#else
__device__ __forceinline__ float sigm_(float x) {
    return 1.0f / (1.0f + __expf(-x));
}
__device__ __forceinline__ float tanh_(float x) {
    float e = __expf(-2.0f * x);
    return (1.0f - e) / (1.0f + e);
}
#endif

__global__ __launch_bounds__(WAVES * 32) void lstm3_wmma_kernel(
    const float* __restrict__ x, float* __restrict__ out,
    const float* __restrict__ wih0, const float* __restrict__ whh0,
    const float* __restrict__ bih0, const float* __restrict__ bhh0,
    const float* __restrict__ wih1, const float* __restrict__ whh1,
    const float* __restrict__ bih1, const float* __restrict__ bhh1,
    const float* __restrict__ wih2, const float* __restrict__ whh2,
    const float* __restrict__ bih2, const float* __restrict__ bhh2)
{
    __shared__ float lds[WAVES * WAVE_F];

    const int tid  = threadIdx.x;
    const int wave = tid >> 5;
    const int lane = tid & 31;
    const int lo   = lane & 15;   // N / M index within half-wave
    const int hi   = lane >> 4;   // half-wave select

    float* chunk = &lds[wave * WAVE_F];   // staged input -> overwritten with h_t (output)
    float* hbuf  = chunk + CHUNK_F;       // h tile carried across chunk boundaries

    const int b0 = (blockIdx.x * WAVES + wave) * 16;  // batch tile base row

    const float* wihL[3] = {wih0, wih1, wih2};
    const float* whhL[3] = {whh0, whh1, whh2};
    const float* bihL[3] = {bih0, bih1, bih2};
    const float* bhhL[3] = {bhh0, bhh1, bhh2};

    for (int layer = 0; layer < 3; ++layer) {
        // ---- B-operand weight fragments, resident in VGPRs for the layer ----
        // B 4x16 layout assumption: VGPR v, lane l -> K = 4s + v + 2*(l>=16), N = l%16.
        // B tile for gate-tile n is W^T[k, 16n+col] = W[16n+col, k].
        float wb[2][4][4][2];  // [0=ih,1=hh][ntile][kstep][vgpr]
        float bias[4];
        {
            const float* Wi = wihL[layer];
            const float* Wh = whhL[layer];
            #pragma unroll
            for (int n = 0; n < 4; ++n) {
                #pragma unroll
                for (int s = 0; s < 4; ++s) {
                    #pragma unroll
                    for (int v = 0; v < 2; ++v) {
                        const int k   = 4 * s + v + 2 * hi;
                        const int row = 16 * n + lo;
                        wb[0][n][s][v] = Wi[row * HD + k];
                        wb[1][n][s][v] = Wh[row * HD + k];
                    }
                }
                bias[n] = bihL[layer][16 * n + lo] + bhhL[layer][16 * n + lo];
            }
        }

        // ---- init state: c in VGPRs (C/D layout), h=0 in LDS ----
        v8f c = {};
        #pragma unroll
        for (int r = 0; r < 8; ++r) {
            const int m = r + 8 * hi;
            hbuf[m * RP + lo] = 0.0f;
        }

        const float* src = (layer == 0) ? x : out;  // previous layer's y lives in d_out

        for (int ch = 0; ch < NCH; ++ch) {
            const int t0 = ch * TC;

            // ---- stage input chunk [TC][16 rows][16] into LDS (coalesced b128) ----
            // 1024 float4 groups; mapping: b = g>>6, t = (g&63)>>2, q = g&3
            // => each instruction touches 512B-contiguous spans per row.
            #pragma unroll 4
            for (int it = 0; it < 32; ++it) {
                const int g = it * 32 + lane;
                const int b = g >> 6;
                const int t = (g & 63) >> 2;
                const int q = g & 3;
                const size_t off = ((size_t)(b0 + b) * T_TOT + (t0 + t)) * HD + q * 4;
                const float4 vv = *(const float4*)(src + off);
                float* dst = chunk + (t * 16 + b) * RP + q * 4;
                ((float2*)dst)[0] = make_float2(vv.x, vv.y);
                ((float2*)dst)[1] = make_float2(vv.z, vv.w);
            }

            // ---- prefetch next chunk into cache (global_prefetch_b8) ----
            if (ch + 1 < NCH) {
                #pragma unroll
                for (int pp = 0; pp < 4; ++pp) {
                    const int g   = pp * 32 + lane;   // 128 segments of 128B = full 16KB chunk
                    const int b   = g >> 3;
                    const int seg = g & 7;
                    __builtin_prefetch(
                        src + ((size_t)(b0 + b) * T_TOT + (t0 + TC)) * HD + seg * 32, 0, 1);
                }
            }

            // ---- recurrence over the chunk ----
            for (int t = 0; t < TC; ++t) {
                float* xrow = chunk + t * 16 * RP;
                // h_{t-1}: slot t-1 of this chunk (it holds the h written last
                // iteration), or hbuf at the chunk boundary.
                const float* hsrc = (t == 0) ? hbuf : (xrow - 16 * RP);

                v8f acc[4];
                #pragma unroll
                for (int n = 0; n < 4; ++n) {
                    v8f a0;
                    #pragma unroll
                    for (int r = 0; r < 8; ++r) a0[r] = bias[n];

                    // x_t @ Wih^T  (A 16x4 layout: VGPR v, lane l -> M=l%16, K=4s+v+2*(l>=16))
                    #pragma unroll
                    for (int s = 0; s < 4; ++s) {
                        v2f af;
                        af.x = xrow[lo * RP + 4 * s + 2 * hi];
                        af.y = xrow[lo * RP + 4 * s + 2 * hi + 1];
                        v2f bf;
                        bf.x = wb[0][n][s][0];
                        bf.y = wb[0][n][s][1];
                        a0 = __builtin_amdgcn_wmma_f32_16x16x4_f32(
                                false, af, false, bf, (short)0, a0, false, false);
                    }
                    // h_{t-1} @ Whh^T
                    #pragma unroll
                    for (int s = 0; s < 4; ++s) {
                        v2f af;
                        af.x = hsrc[lo * RP + 4 * s + 2 * hi];
                        af.y = hsrc[lo * RP + 4 * s + 2 * hi + 1];
                        v2f bf;
                        bf.x = wb[1][n][s][0];
                        bf.y = wb[1][n][s][1];
                        a0 = __builtin_amdgcn_wmma_f32_16x16x4_f32(
                                false, af, false, bf, (short)0, a0, false, false);
                    }
                    acc[n] = a0;
                }

                // ---- gate nonlinearities + state update (C/D layout, per element) ----
                v8f hnew = {};
                #pragma unroll
                for (int r = 0; r < 8; ++r) {
                    const float ig = sigm_(acc[0][r]);
                    const float fg = sigm_(acc[1][r]);
                    const float gg = tanh_(acc[2][r]);
                    const float og = sigm_(acc[3][r]);
                    const float cn = fg * c[r] + ig * gg;
                    c[r]    = cn;
                    hnew[r] = og * tanh_(cn);
                }

                // D-layout -> row-major LDS into output slot t (x_t already
                // consumed above; in-wave DS ordering makes this safe). The
                // next timestep reads its h from this slot; only the chunk
                // boundary needs hbuf.
                #pragma unroll
                for (int r = 0; r < 8; ++r) {
                    const int m = r + 8 * hi;
                    xrow[m * RP + lo] = hnew[r];
                }
                if (t == TC - 1) {
                    #pragma unroll
                    for (int r = 0; r < 8; ++r) {
                        const int m = r + 8 * hi;
                        hbuf[m * RP + lo] = hnew[r];
                    }
                }
            }

            // ---- write chunk of h back to out (+ residual x on final layer) ----
            const bool last = (layer == 2);
            #pragma unroll 4
            for (int it = 0; it < 32; ++it) {
                const int g = it * 32 + lane;
                const int b = g >> 6;
                const int t = (g & 63) >> 2;
                const int q = g & 3;
                const float* sp = chunk + (t * 16 + b) * RP + q * 4;
                const float2 p0 = ((const float2*)sp)[0];
                const float2 p1 = ((const float2*)sp)[1];
                float4 vv = make_float4(p0.x, p0.y, p1.x, p1.y);
                const size_t off = ((size_t)(b0 + b) * T_TOT + (t0 + t)) * HD + q * 4;
                if (last) {
                    const float4 xr = *(const float4*)(x + off);
                    vv.x += xr.x; vv.y += xr.y; vv.z += xr.z; vv.w += xr.w;
                }
                *(float4*)(out + off) = vv;
            }
        }
    }
}

extern "C" void kernel_launch(void* const* d_in, const int* in_sizes, int n_in,
                              void* d_out, int out_size, void* d_ws, size_t ws_size,
                              hipStream_t stream) {
    (void)in_sizes; (void)n_in; (void)out_size; (void)d_ws; (void)ws_size;
    const float* x    = (const float*)d_in[0];
    const float* wih0 = (const float*)d_in[1];
    const float* whh0 = (const float*)d_in[2];
    const float* bih0 = (const float*)d_in[3];
    const float* bhh0 = (const float*)d_in[4];
    const float* wih1 = (const float*)d_in[5];
    const float* whh1 = (const float*)d_in[6];
    const float* bih1 = (const float*)d_in[7];
    const float* bhh1 = (const float*)d_in[8];
    const float* wih2 = (const float*)d_in[9];
    const float* whh2 = (const float*)d_in[10];
    const float* bih2 = (const float*)d_in[11];
    const float* bhh2 = (const float*)d_in[12];
    float* out = (float*)d_out;

    dim3 grid(B_TOT / (16 * WAVES));  // 128 workgroups
    dim3 block(32 * WAVES);           // 2 independent wave32 tiles per WG

    hipLaunchKernelGGL(lstm3_wmma_kernel, grid, block, 0, stream,
                       x, out,
                       wih0, whh0, bih0, bhh0,
                       wih1, whh1, bih1, bhh1,
                       wih2, whh2, bih2, bhh2);
}